// Decoder_85341000171629
// MI455X (gfx1250) — compile-verified
//
#include <hip/hip_runtime.h>
#include <cstdint>
#include <cstddef>

// ---------------------------------------------------------------------------
// Compile-time replication of np.random.default_rng(0).permutation(...)
// (SeedSequence -> PCG64 XSL-RR 128/64 -> masked-rejection Fisher-Yates)
// ---------------------------------------------------------------------------
typedef unsigned __int128 u128;

struct NpTables { int idx1[16]; int idx2[128]; int idx3[512]; };

constexpr u128 mk128(uint64_t hi, uint64_t lo) { return ((u128)hi << 64) | (u128)lo; }
constexpr u128 PCG_MULT = mk128(2549297995355413924ULL, 4865540595714422341ULL);

struct PcgState { u128 state; u128 inc; uint32_t buf; bool has; };

constexpr uint32_t np_hashmix(uint32_t v, uint32_t& hc) {
    v ^= hc; hc *= 0x931e8875u; v *= hc; v ^= v >> 16; return v;
}
constexpr uint32_t np_mix(uint32_t x, uint32_t y) {
    uint32_t r = (x * 0xca01f9ddu) ^ (y * 0x4973f715u); r ^= r >> 16; return r;
}
constexpr uint64_t pcg_next64(PcgState& g) {
    g.state = g.state * PCG_MULT + g.inc;
    uint64_t hi = (uint64_t)(g.state >> 64);
    uint64_t lo = (uint64_t)g.state;
    unsigned rot = (unsigned)(g.state >> 122);
    uint64_t x = hi ^ lo;
    return rot ? ((x >> rot) | (x << (64u - rot))) : x;
}
constexpr uint32_t pcg_next32(PcgState& g) {
    if (g.has) { g.has = false; return g.buf; }
    uint64_t n = pcg_next64(g);
    g.has = true; g.buf = (uint32_t)(n >> 32);
    return (uint32_t)n;
}
constexpr uint32_t rand_interval(PcgState& g, uint32_t mx) {
    if (mx == 0u) return 0u;
    uint32_t mask = mx;
    mask |= mask >> 1; mask |= mask >> 2; mask |= mask >> 4;
    mask |= mask >> 8; mask |= mask >> 16;
    uint32_t v = 0u;
    while ((v = (pcg_next32(g) & mask)) > mx) { }
    return v;
}
constexpr NpTables make_tables() {
    uint32_t pool[4] = {0, 0, 0, 0};
    uint32_t hc = 0x43b0d7e5u;
    for (int i = 0; i < 4; ++i) pool[i] = np_hashmix(0u, hc);
    for (int s = 0; s < 4; ++s)
        for (int t = 0; t < 4; ++t)
            if (s != t) pool[t] = np_mix(pool[t], np_hashmix(pool[s], hc));
    uint32_t st[8] = {};
    uint32_t hb = 0x8b51f9ddu;
    for (int i = 0; i < 8; ++i) {
        uint32_t v = pool[i & 3];
        v ^= hb; hb *= 0x58f38dedu; v *= hb; v ^= v >> 16;
        st[i] = v;
    }
    uint64_t v0 = (uint64_t)st[0] | ((uint64_t)st[1] << 32);
    uint64_t v1 = (uint64_t)st[2] | ((uint64_t)st[3] << 32);
    uint64_t v2 = (uint64_t)st[4] | ((uint64_t)st[5] << 32);
    uint64_t v3 = (uint64_t)st[6] | ((uint64_t)st[7] << 32);
    u128 initstate = mk128(v0, v1);
    u128 initseq   = mk128(v2, v3);
    PcgState g = {0, 0, 0, false};
    g.inc = (initseq << 1) | (u128)1;
    g.state = g.state * PCG_MULT + g.inc;
    g.state += initstate;
    g.state = g.state * PCG_MULT + g.inc;
    int arr[656] = {};
    for (int i = 0; i < 656; ++i) arr[i] = (i < 16) ? 1 : (i < 144) ? 2 : 3;
    for (int i = 655; i >= 1; --i) {
        int j = (int)rand_interval(g, (uint32_t)i);
        int t = arr[i]; arr[i] = arr[j]; arr[j] = t;
    }
    NpTables T = {};
    int n1 = 0, n2 = 0, n3 = 0;
    for (int p = 0; p < 656; ++p) {
        if (arr[p] == 1) T.idx1[n1++] = p;
        else if (arr[p] == 2) T.idx2[n2++] = p;
        else T.idx3[n3++] = p;
    }
    return T;
}
__constant__ NpTables g_tbl = make_tables();

// ---------------------------------------------------------------------------
// Kernels
// ---------------------------------------------------------------------------
typedef float v2f __attribute__((ext_vector_type(2)));
typedef float v8f __attribute__((ext_vector_type(8)));

#define HH 512
#define BB 64
#define LL 656
#define KC 64                       // K-chunk per LDS stage
#define LDSP 68                     // padded row stride (floats); rows 16B aligned
#define BUF_FLOATS (64 * LDSP)      // 4352 floats per A (or B) tile
#define PAIR_FLOATS (2 * BUF_FLOATS)
#define PAIR_BYTES (PAIR_FLOATS * 4)   // 34816 B per double-buffer stage
#define B_OFF_BYTES (BUF_FLOATS * 4)   // 17408 B
#define SMEM_BYTES (2 * PAIR_BYTES)    // 69632 B dynamic LDS (WGP has 320 KB)

typedef __attribute__((address_space(3))) void lds_void_t;
__device__ __forceinline__ uint32_t lds_lo(void* p) {
    return (uint32_t)(uintptr_t)(lds_void_t*)p;
}

__global__ __launch_bounds__(256) void zero_ws(float* __restrict__ p, int n) {
    int i = blockIdx.x * 256 + threadIdx.x;
    if (i < n) p[i] = 0.0f;
}

// GEMM1: X_d[64,512] += G_d[64, c*512] @ W1_d^T, split-K with fp32 atomics.
// Double-buffered GLOBAL_LOAD_ASYNC_TO_LDS_B128 pipeline feeding f32 WMMA.
__global__ __launch_bounds__(256) void gemm1_wmma(
    const float* __restrict__ emb,
    const float* __restrict__ W1a, const float* __restrict__ W1b,
    const float* __restrict__ W1c, float* __restrict__ Xws)
{
    extern __shared__ float smem[];

    int bid = blockIdx.x;
    int d, local;
    if (bid < 32)       { d = 0; local = bid; }
    else if (bid < 160) { d = 1; local = bid - 32; }
    else                { d = 2; local = bid - 160; }

    const int S      = (d == 0) ? 4 : (d == 1) ? 16 : 64;
    const int c      = (d == 0) ? 16 : (d == 1) ? 128 : 512;
    const float* W1  = (d == 0) ? W1a : (d == 1) ? W1b : W1c;
    const int* idx   = (d == 0) ? g_tbl.idx1 : (d == 1) ? g_tbl.idx2 : g_tbl.idx3;

    const int nt   = local / S;
    const int sp   = local % S;
    const size_t Kdim = (size_t)c * HH;
    const int Kper = (int)(Kdim / (size_t)S);   // 2048 or 4096
    const int k0   = sp * Kper;
    const int h0   = nt * 64;

    const int tid   = threadIdx.x;
    const int wave  = tid >> 5;
    const int lane  = tid & 31;
    const int mt    = wave & 3;            // batch-row tile (16 rows)
    const int ng    = wave >> 2;           // column group (0/1)
    const int nn0   = ng * 32;
    const int nn1   = ng * 32 + 16;
    const int lhalf = lane >> 4;
    const int l15   = lane & 15;

    // Per-lane constants for async staging: 8 b128 DMAs per wave per chunk
    // (4 for the gathered emb tile A, 4 for the W1 tile B).
    uint32_t rq[4];
    uint64_t aoff[4], boff[4];
    #pragma unroll
    for (int v = 0; v < 4; ++v) {
        int flat = (wave * 4 + v) * 32 + lane;   // 0..1023
        int row  = flat >> 4;                    // 0..63
        int q    = flat & 15;                    // float4 slot in row
        rq[v]   = (uint32_t)(row * (LDSP * 4) + q * 16);                 // LDS bytes
        aoff[v] = ((uint64_t)row * (LL * HH) + (uint64_t)(q * 4)) * 4;   // emb bytes
        boff[v] = ((uint64_t)(h0 + row) * Kdim + (uint64_t)(q * 4)) * 4; // W1 bytes
    }
    const uint32_t smem_lds = lds_lo(smem);
    const uint32_t ldsA0 = smem_lds;
    const uint32_t ldsA1 = smem_lds + PAIR_BYTES;

    auto issue = [&](int chunk, uint32_t lbA) {
        const int kc  = k0 + chunk * KC;
        const int j   = kc >> 9;               // KC divides 512: single label
        const int ko  = kc & 511;
        const int lab = idx[j];
        const uint64_t baseA = (uint64_t)(uintptr_t)emb +
                               ((uint64_t)lab * HH + (uint64_t)ko) * 4;
        const uint64_t baseB = (uint64_t)(uintptr_t)W1 + (uint64_t)kc * 4;
        const uint32_t lbB = lbA + B_OFF_BYTES;
        #pragma unroll
        for (int v = 0; v < 4; ++v) {
            uint64_t ga = baseA + aoff[v];
            uint64_t gb = baseB + boff[v];
            asm volatile("global_load_async_to_lds_b128 %0, %1, off"
                         :: "v"(lbA + rq[v]), "v"(ga) : "memory");
            asm volatile("global_load_async_to_lds_b128 %0, %1, off"
                         :: "v"(lbB + rq[v]), "v"(gb) : "memory");
        }
    };

    v8f acc0 = {}; v8f acc1 = {};
    const int arow  = mt * 16 + l15;
    const int kbase = lhalf * 2;

    const int nchunks = Kper / KC;             // 32 or 64
    issue(0, ldsA0);
    issue(1, ldsA1);

    for (int i = 0; i < nchunks; ++i) {
        // Async loads complete in order: <=8 outstanding leaves only chunk i+1
        if (i + 1 < nchunks) asm volatile("s_wait_asynccnt 0x8" ::: "memory");
        else                 asm volatile("s_wait_asynccnt 0x0" ::: "memory");
        __syncthreads();                       // publish LDS across all waves

        const float* Af = smem + (size_t)(i & 1) * PAIR_FLOATS;
        const float* Bf = Af + BUF_FLOATS;
        #pragma unroll
        for (int ks = 0; ks < KC; ks += 4) {
            v2f a  = *(const v2f*)(Af + arow * LDSP + ks + kbase);
            v2f b0 = *(const v2f*)(Bf + (nn0 + l15) * LDSP + ks + kbase);
            v2f b1 = *(const v2f*)(Bf + (nn1 + l15) * LDSP + ks + kbase);
            acc0 = __builtin_amdgcn_wmma_f32_16x16x4_f32(
                false, a, false, b0, (short)0, acc0, false, false);
            acc1 = __builtin_amdgcn_wmma_f32_16x16x4_f32(
                false, a, false, b1, (short)0, acc1, false, false);
        }
        __syncthreads();                       // all waves done reading buf i&1
        if (i + 2 < nchunks) issue(i + 2, (i & 1) ? ldsA1 : ldsA0);
    }

    float* Xd = Xws + (size_t)d * BB * HH;
    const int col0 = h0 + nn0 + l15;
    const int col1 = h0 + nn1 + l15;
    #pragma unroll
    for (int i = 0; i < 8; ++i) {
        int brow = mt * 16 + i + lhalf * 8;
        unsafeAtomicAdd(Xd + (size_t)brow * HH + col0, acc0[i]);
        unsafeAtomicAdd(Xd + (size_t)brow * HH + col1, acc1[i]);
    }
}

// GEMM2 + bias + scatter: out[b, idx_d[ci]] = X_d[b,:] . Wp_d[ci,:] + bp_d[ci]
__global__ __launch_bounds__(256) void proj_scatter(
    const float* __restrict__ Xws,
    const float* __restrict__ Wp1, const float* __restrict__ Wp2,
    const float* __restrict__ Wp3,
    const float* __restrict__ bp1, const float* __restrict__ bp2,
    const float* __restrict__ bp3,
    float* __restrict__ out)
{
    int tid = blockIdx.x * 256 + threadIdx.x;
    if (tid >= BB * LL) return;
    int p = tid / BB;
    int b = tid % BB;
    int d, ci;
    const float* Wp; const float* bp; const int* idx;
    if (p < 16)       { d = 0; ci = p;       Wp = Wp1; bp = bp1; idx = g_tbl.idx1; }
    else if (p < 144) { d = 1; ci = p - 16;  Wp = Wp2; bp = bp2; idx = g_tbl.idx2; }
    else              { d = 2; ci = p - 144; Wp = Wp3; bp = bp3; idx = g_tbl.idx3; }

    const float4* xr = (const float4*)(Xws + ((size_t)d * BB + b) * HH);
    const float4* wr = (const float4*)(Wp + (size_t)ci * HH);
    float s = 0.0f;
    #pragma unroll 4
    for (int k = 0; k < HH / 4; ++k) {
        float4 x = xr[k], w = wr[k];
        s += x.x * w.x + x.y * w.y + x.z * w.z + x.w * w.w;
    }
    out[(size_t)b * LL + idx[ci]] = s + bp[ci];
}

// ---------------------------------------------------------------------------
// Launch
// ---------------------------------------------------------------------------
extern "C" void kernel_launch(void* const* d_in, const int* in_sizes, int n_in,
                              void* d_out, int out_size, void* d_ws, size_t ws_size,
                              hipStream_t stream)
{
    const float* emb  = (const float*)d_in[0];
    // d_in[1] = proj_label_repr (unused by reference)
    const float* W1_1 = (const float*)d_in[2];
    const float* Wp_1 = (const float*)d_in[3];
    const float* bp_1 = (const float*)d_in[4];
    const float* W1_2 = (const float*)d_in[5];
    const float* Wp_2 = (const float*)d_in[6];
    const float* bp_2 = (const float*)d_in[7];
    const float* W1_3 = (const float*)d_in[8];
    const float* Wp_3 = (const float*)d_in[9];
    const float* bp_3 = (const float*)d_in[10];

    float* Xws = (float*)d_ws;              // 3 * 64 * 512 floats = 393 KB
    float* out = (float*)d_out;             // [64, 656]

    const int xn = 3 * BB * HH;
    zero_ws<<<(xn + 255) / 256, 256, 0, stream>>>(Xws, xn);
    gemm1_wmma<<<672, 256, SMEM_BYTES, stream>>>(emb, W1_1, W1_2, W1_3, Xws);
    proj_scatter<<<(BB * LL + 255) / 256, 256, 0, stream>>>(
        Xws, Wp_1, Wp_2, Wp_3, bp_1, bp_2, bp_3, out);
}